// GCN_8701603741946
// MI455X (gfx1250) — compile-verified
//
#include <hip/hip_runtime.h>
#include <hip/hip_bf16.h>

typedef __attribute__((ext_vector_type(2))) float v2f;
typedef __attribute__((ext_vector_type(8))) float v8f;

// ---------------------------------------------------------------------------
// Degree / normalization
// ---------------------------------------------------------------------------
__global__ __launch_bounds__(256) void deg_init_kernel(float* __restrict__ deg, int n) {
    int i = blockIdx.x * blockDim.x + threadIdx.x;
    if (i < n) deg[i] = 1.0f;   // self-loop contributes 1 to every node's degree
}

__global__ __launch_bounds__(256) void deg_accum_kernel(const int* __restrict__ col,
                                                        float* __restrict__ deg, int e_count) {
    int e = blockIdx.x * blockDim.x + threadIdx.x;
    if (e < e_count) atomicAdd(&deg[col[e]], 1.0f);
}

__global__ __launch_bounds__(256) void dinv_kernel(float* __restrict__ deg, int n) {
    int i = blockIdx.x * blockDim.x + threadIdx.x;
    if (i < n) deg[i] = rsqrtf(deg[i]);   // deg >= 1 always (self-loops)
}

// ---------------------------------------------------------------------------
// h1 = x[N,128] @ W1[128,4] via V_WMMA_F32_16X16X4_F32 (one wave per 16-row tile)
//
// A (16x4 f32): lane L<16 holds row L, K=k0+{0,1}; lane L>=16 holds row L-16,
//               K=k0+{2,3}  (ISA 7.12.2, 32-bit A 16x4 layout).
// B (4x16 f32): lane holds col N=lane&15; VGPR0=K=(lane<16?0:2), VGPR1=K+1.
//               Columns N>=4 are forced to 0 (W1 has only 4 output cols) with a
//               select (no EXEC divergence: WMMA requires EXEC all ones).
// D (16x16 f32, 8 VGPRs): VGPR v -> row v (lanes 0-15) / row v+8 (lanes 16-31),
//               col = lane&15. Lanes with col<4 store the useful columns.
// ---------------------------------------------------------------------------
__global__ __launch_bounds__(256) void gemm_x_w1_kernel(const float* __restrict__ x,
                                                        const float* __restrict__ W1,
                                                        float* __restrict__ h1,
                                                        int n, int n_tiles) {
    int gwave = (blockIdx.x * 256 + threadIdx.x) >> 5;   // global wave id == tile id
    int lane  = threadIdx.x & 31;
    if (gwave >= n_tiles) return;

    int m     = lane & 15;             // A: row-in-tile, B/D: column
    int khalf = (lane >> 4) << 1;      // 0 for lanes 0-15, 2 for lanes 16-31

    int row_g = gwave * 16 + m;
    int row_c = row_g < n ? row_g : (n - 1);             // clamp loads (guard stores)
    const float* xrow = x + (size_t)row_c * 128 + khalf;

    float sel = (m < 4) ? 1.0f : 0.0f;                   // zero out B columns >= 4
    int   nb  = m & 3;

    v8f acc = {};
#pragma unroll
    for (int k0 = 0; k0 < 128; k0 += 4) {
        v2f a;
        a.x = xrow[k0];
        a.y = xrow[k0 + 1];
        v2f b;
        b.x = W1[(size_t)(k0 + khalf) * 4 + nb] * sel;
        b.y = W1[(size_t)(k0 + khalf + 1) * 4 + nb] * sel;
        acc = __builtin_amdgcn_wmma_f32_16x16x4_f32(
            /*neg_a=*/false, a, /*neg_b=*/false, b,
            /*c_mod=*/(short)0, acc, /*reuse_a=*/false, /*reuse_b=*/false);
    }

    if (m < 4) {
        int rbase = gwave * 16 + ((lane >> 4) << 3);     // +8 rows for upper half
#pragma unroll
        for (int v = 0; v < 8; ++v) {
            int r = rbase + v;
            if (r < n) h1[(size_t)r * 4 + m] = acc[v];
        }
    }
}

// ---------------------------------------------------------------------------
// Propagation: acc[i] = sum_{e: col=i} h[row]*dinv[row]*dinv[i]
//                     + h[i]*dinv[i]^2 (self-loop) + bias
// ---------------------------------------------------------------------------
template <int D>
__global__ __launch_bounds__(256) void acc_init_kernel(const float* __restrict__ hraw,
                                                       const float* __restrict__ dinv,
                                                       const float* __restrict__ bias,
                                                       float* __restrict__ acc, int n) {
    int i = blockIdx.x * blockDim.x + threadIdx.x;
    if (i >= n) return;
    float di  = dinv[i];
    float di2 = di * di;
#pragma unroll
    for (int k = 0; k < D; ++k)
        acc[(size_t)i * D + k] = hraw[(size_t)i * D + k] * di2 + bias[k];
}

template <int D>
__global__ __launch_bounds__(256) void edge_prop_kernel(const int* __restrict__ row,
                                                        const int* __restrict__ col,
                                                        const float* __restrict__ hraw,
                                                        const float* __restrict__ dinv,
                                                        float* acc, int e_count) {
    int e = blockIdx.x * blockDim.x + threadIdx.x;
    if (e >= e_count) return;
    int r = row[e];
    int c = col[e];
    float w = dinv[r] * dinv[c];                 // norm, recomputed from L2-resident dinv
    const float* hp = hraw + (size_t)r * D;      // 16B/8B L2-resident gather
    float*       ap = acc  + (size_t)c * D;
#pragma unroll
    for (int k = 0; k < D; ++k)
        atomicAdd(&ap[k], hp[k] * w);            // global_atomic_add_f32 (no return)
}

// ---------------------------------------------------------------------------
// Finalize layers: h_next = tanh(acc) @ W  (tiny dense projections, pure VALU)
// ---------------------------------------------------------------------------
__global__ __launch_bounds__(256) void finalize12_kernel(const float* __restrict__ acc,
                                                         const float* __restrict__ W2,
                                                         float* __restrict__ h2, int n) {
    int i = blockIdx.x * blockDim.x + threadIdx.x;
    if (i >= n) return;
    float t[4];
#pragma unroll
    for (int k = 0; k < 4; ++k) t[k] = tanhf(acc[(size_t)i * 4 + k]);
#pragma unroll
    for (int j = 0; j < 4; ++j) {
        float s = 0.0f;
#pragma unroll
        for (int k = 0; k < 4; ++k) s += t[k] * W2[k * 4 + j];
        h2[(size_t)i * 4 + j] = s;
    }
}

__global__ __launch_bounds__(256) void finalize23_kernel(const float* __restrict__ acc,
                                                         const float* __restrict__ W3,
                                                         float* __restrict__ h3, int n) {
    int i = blockIdx.x * blockDim.x + threadIdx.x;
    if (i >= n) return;
    float t[4];
#pragma unroll
    for (int k = 0; k < 4; ++k) t[k] = tanhf(acc[(size_t)i * 4 + k]);
#pragma unroll
    for (int j = 0; j < 2; ++j) {
        float s = 0.0f;
#pragma unroll
        for (int k = 0; k < 4; ++k) s += t[k] * W3[k * 2 + j];
        h3[(size_t)i * 2 + j] = s;
    }
}

__global__ __launch_bounds__(256) void finalize_out_kernel(const float* __restrict__ acc,
                                                           const float* __restrict__ Wc,
                                                           const float* __restrict__ bc,
                                                           float* __restrict__ out,
                                                           float* __restrict__ hout, int n) {
    int i = blockIdx.x * blockDim.x + threadIdx.x;
    if (i >= n) return;
    float t0 = tanhf(acc[(size_t)i * 2 + 0]);
    float t1 = tanhf(acc[(size_t)i * 2 + 1]);
    hout[(size_t)i * 2 + 0] = t0;
    hout[(size_t)i * 2 + 1] = t1;
#pragma unroll
    for (int c = 0; c < 16; ++c)
        out[(size_t)i * 16 + c] = t0 * Wc[c] + t1 * Wc[16 + c] + bc[c];
}

// ---------------------------------------------------------------------------
// Launcher
// ---------------------------------------------------------------------------
extern "C" void kernel_launch(void* const* d_in, const int* in_sizes, int n_in,
                              void* d_out, int out_size, void* d_ws, size_t ws_size,
                              hipStream_t stream) {
    const float* x   = (const float*)d_in[0];
    const int*   ei  = (const int*)d_in[1];
    const float* W1  = (const float*)d_in[2];
    const float* b1  = (const float*)d_in[3];
    const float* W2  = (const float*)d_in[4];
    const float* b2  = (const float*)d_in[5];
    const float* W3  = (const float*)d_in[6];
    const float* b3  = (const float*)d_in[7];
    const float* Wc  = (const float*)d_in[8];
    const float* bc  = (const float*)d_in[9];

    const int N = in_sizes[0] / 128;
    const int E = in_sizes[1] / 2;
    const int* row = ei;          // edge_index[0]
    const int* col = ei + E;      // edge_index[1]

    float* out_logits = (float*)d_out;               // [N,16]
    float* out_h      = (float*)d_out + (size_t)N * 16;  // [N,2]

    // Workspace: dinv[N] | h[4N] | acc[4N]  (36 MB at N=1M)
    float* dinv = (float*)d_ws;
    float* hbuf = dinv + (size_t)N;
    float* accb = dinv + (size_t)5 * N;

    const int T  = 256;
    const int gN = (N + T - 1) / T;
    const int gE = (E + T - 1) / T;

    // 1) degree (self-loops folded into init) -> dinv = rsqrt(deg)
    deg_init_kernel<<<gN, T, 0, stream>>>(dinv, N);
    deg_accum_kernel<<<gE, T, 0, stream>>>(col, dinv, E);
    dinv_kernel<<<gN, T, 0, stream>>>(dinv, N);

    // 2) h1 = x @ W1  (WMMA fp32)
    const int n_tiles = (N + 15) / 16;
    const int gG = (n_tiles + 7) / 8;                // 8 waves of 32 per block
    gemm_x_w1_kernel<<<gG, T, 0, stream>>>(x, W1, hbuf, N, n_tiles);

    // 3) layer 1 propagate + tanh + project (W2)
    acc_init_kernel<4><<<gN, T, 0, stream>>>(hbuf, dinv, b1, accb, N);
    edge_prop_kernel<4><<<gE, T, 0, stream>>>(row, col, hbuf, dinv, accb, E);
    finalize12_kernel<<<gN, T, 0, stream>>>(accb, W2, hbuf, N);

    // 4) layer 2 propagate + tanh + project (W3)
    acc_init_kernel<4><<<gN, T, 0, stream>>>(hbuf, dinv, b2, accb, N);
    edge_prop_kernel<4><<<gE, T, 0, stream>>>(row, col, hbuf, dinv, accb, E);
    finalize23_kernel<<<gN, T, 0, stream>>>(accb, W3, hbuf, N);

    // 5) layer 3 propagate + tanh -> h3, then classifier -> out
    acc_init_kernel<2><<<gN, T, 0, stream>>>(hbuf, dinv, b3, accb, N);
    edge_prop_kernel<2><<<gE, T, 0, stream>>>(row, col, hbuf, dinv, accb, E);
    finalize_out_kernel<<<gN, T, 0, stream>>>(accb, Wc, bc, out_logits, out_h, N);
}